// ADMM_32246614459005
// MI455X (gfx1250) — compile-verified
//
#include <hip/hip_runtime.h>
#include <math.h>

#define B_ 4096
#define M_ 100
#define N_ 80
#define L_ 10
#define THREADS 256

typedef __attribute__((ext_vector_type(2))) float v2f;
typedef __attribute__((ext_vector_type(8))) float v8f;

// Zero the single loss accumulator slot (d_out[B*N]) before the main kernel
// accumulates into it with global atomics. Separate kernel on the same stream
// => ordered, graph-capture safe.
__global__ void __launch_bounds__(64)
admm_zero_loss(float* out) {
    if (threadIdx.x == 0) out[B_ * N_] = 0.0f;
}

__global__ void __launch_bounds__(THREADS)
admm_kernel(const float* __restrict__ x_ini,
            const float* __restrict__ y,
            const float* __restrict__ H,
            const float* __restrict__ xt,
            const float* __restrict__ ILbar,
            const float* __restrict__ tau,
            const float* __restrict__ gamma1,
            const float* __restrict__ gamma3,
            float* __restrict__ out) {
    __shared__ float sH[M_ * N_];   // H for this batch, row-major [m][n]  (32 KB)
    __shared__ float sA[N_ * N_];   // HTH + tau*I -> in-place inverse     (25.6 KB)
    __shared__ float sHTy[N_];
    __shared__ float s_y[M_];
    __shared__ float s_colk[N_];
    __shared__ float s_rhs[N_];
    __shared__ float s_x[N_];
    __shared__ float s_z[N_];
    __shared__ float s_u[N_];
    __shared__ float s_lsum;

    const int tid = threadIdx.x;
    const int b   = blockIdx.x;
    const float* __restrict__ Hb = H + (size_t)b * (M_ * N_);

    // ---- stage H (128-bit loads: 2000 float4), y into LDS; init state ----
    {
        const float4* __restrict__ Hb4 = reinterpret_cast<const float4*>(Hb);
        float4* sH4 = reinterpret_cast<float4*>(sH);
        for (int idx = tid; idx < (M_ * N_) / 4; idx += THREADS)
            sH4[idx] = Hb4[idx];                 // global_load_b128 -> ds_store_b128
    }
    if (tid < M_) s_y[tid] = y[b * M_ + tid];
    if (tid < N_) {
        float xi = x_ini[b * N_ + tid];
        s_x[tid] = xi;
        s_z[tid] = xi;
        s_u[tid] = 0.0f;
    }
    if (tid == 0) s_lsum = 0.0f;
    __syncthreads();

    // ---- HTy[n] = sum_m H[m][n] * y[m] ----
    if (tid < N_) {
        float s = 0.0f;
        for (int m = 0; m < M_; ++m) s += sH[m * N_ + tid] * s_y[m];
        sHTy[tid] = s;
    }

    const int wave  = tid >> 5;          // 8 waves of 32
    const int lane  = tid & 31;
    const int rN    = lane & 15;         // row/col within 16
    const int khalf = (lane >> 4) * 2;   // K base: 0 (lanes 0-15) or 2 (lanes 16-31)
    const int mrow  = (lane >> 4) * 8;   // C/D row offset for upper half-wave

    float tau_prev = 0.0f;

    for (int ii = 0; ii < L_; ++ii) {
        const float tau_i = tau[ii];             // uniform scalar loads
        const float ilb   = ILbar[ii];
        const float g1    = 0.5f / gamma1[ii];
        const float g3    = 0.5f / gamma3[ii];

        if (ii == 0 || tau_i != tau_prev) {      // wave-uniform branch
            __syncthreads();                     // prior readers of sA done
            // ---- HTH via V_WMMA_F32_16X16X4_F32: 5x5 tiles, K = 100 in 25 steps ----
            for (int tile = wave; tile < 25; tile += 8) {
                const int i0 = (tile / 5) * 16;
                const int j0 = (tile % 5) * 16;
                v8f acc = {0.f, 0.f, 0.f, 0.f, 0.f, 0.f, 0.f, 0.f};
                for (int m0 = 0; m0 < M_; m0 += 4) {
                    v2f af, bf;
                    // A-frag: (HT)[i0+rN][m0+khalf .. +1] = H[m0+k][i0+rN]
                    af.x = sH[(m0 + khalf)     * N_ + (i0 + rN)];
                    af.y = sH[(m0 + khalf + 1) * N_ + (i0 + rN)];
                    // B-frag: H[m0+k][j0+rN]
                    bf.x = sH[(m0 + khalf)     * N_ + (j0 + rN)];
                    bf.y = sH[(m0 + khalf + 1) * N_ + (j0 + rN)];
                    acc = __builtin_amdgcn_wmma_f32_16x16x4_f32(
                        false, af, false, bf, (short)0, acc, false, false);
                }
#pragma unroll
                for (int v = 0; v < 8; ++v) {
                    sA[(i0 + v + mrow) * N_ + (j0 + rN)] = acc[v];
                }
            }
            __syncthreads();
            // ---- A = HTH + tau*I ----
            if (tid < N_) sA[tid * N_ + tid] += tau_i;
            __syncthreads();
            // ---- in-place Gauss-Jordan inversion (SPD + tau*I: no pivoting) ----
            for (int k = 0; k < N_; ++k) {
                const float pinv = 1.0f / sA[k * N_ + k];
                if (tid < N_) s_colk[tid] = sA[tid * N_ + k]; // snapshot col k
                __syncthreads();
                if (tid < N_) {                               // scale pivot row
                    const int j = tid;
                    sA[k * N_ + j] = (j == k) ? pinv : sA[k * N_ + j] * pinv;
                }
                __syncthreads();
                for (int idx = tid; idx < N_ * N_; idx += THREADS) {
                    const int i = idx / N_;
                    const int j = idx - i * N_;
                    if (i != k) {
                        const float f = s_colk[i];
                        sA[idx] = (j == k) ? (-f * pinv)
                                           : (sA[idx] - f * sA[k * N_ + j]);
                    }
                }
                __syncthreads();
            }
        }
        tau_prev = tau_i;

        // ---- rhs = HTy + u + tau*z ----
        if (tid < N_) s_rhs[tid] = sHTy[tid] + s_u[tid] + tau_i * s_z[tid];
        __syncthreads();
        // ---- x = Ainv @ rhs ; elementwise z/u updates ----
        if (tid < N_) {
            float s = 0.0f;
            for (int j = 0; j < N_; ++j) s += sA[tid * N_ + j] * s_rhs[j];
            s_x[tid] = s;
            const float t  = s - s_u[tid] * ilb;
            const float zv = tanhf(t * g1) + tanhf((t - 2.0f) * g3)
                           + tanhf((t + 2.0f) * g3);
            s_z[tid] = zv;
            s_u[tid] = s_u[tid] + tau_i * (zv - s);
        }
        __syncthreads();
    }

    // ---- outputs: x and loss = sum(dx^2) + sum(dz^2) ----
    if (tid < N_) {
        const float xv  = s_x[tid];
        const float zv  = s_z[tid];
        const float xtv = xt[b * N_ + tid];
        out[b * N_ + tid] = xv;
        const float dx = xv - xtv;
        const float dz = zv - xtv;
        atomicAdd(&s_lsum, dx * dx + dz * dz);   // ds_add_f32
    }
    __syncthreads();
    if (tid == 0) atomicAdd(out + B_ * N_, s_lsum);  // global_atomic_add_f32
}

extern "C" void kernel_launch(void* const* d_in, const int* in_sizes, int n_in,
                              void* d_out, int out_size, void* d_ws, size_t ws_size,
                              hipStream_t stream) {
    (void)in_sizes; (void)n_in; (void)d_ws; (void)ws_size; (void)out_size;
    const float* x_ini  = (const float*)d_in[0];
    const float* y      = (const float*)d_in[1];
    const float* H      = (const float*)d_in[2];
    const float* xt     = (const float*)d_in[3];
    const float* ILbar  = (const float*)d_in[4];
    const float* tau    = (const float*)d_in[5];
    const float* gamma1 = (const float*)d_in[6];
    const float* gamma3 = (const float*)d_in[7];
    float* out = (float*)d_out;

    admm_zero_loss<<<1, 64, 0, stream>>>(out);
    admm_kernel<<<B_, THREADS, 0, stream>>>(x_ini, y, H, xt,
                                            ILbar, tau, gamma1, gamma3, out);
}